// HoughVotingLayer_86440511799559
// MI455X (gfx1250) — compile-verified
//
#include <hip/hip_runtime.h>

typedef float v2f __attribute__((ext_vector_type(2)));
typedef float v8f __attribute__((ext_vector_type(8)));

#define BATCH 32
#define CH    2
#define IMH   224
#define IMW   224
#define NPTS  4096
#define NPAIR 256
#define THREADS 256
#define NTILES (NPTS / 16)   // 256 P-tiles

// One workgroup per batch instance. wave32: 8 waves / block.
__global__ __launch_bounds__(THREADS) void hough_voting_kernel(
    const float* __restrict__ uv_img,   // (B, 2, H, W) fp32
    const int*   __restrict__ pts,      // (B, P, 2) int32  [row, col]
    const int*   __restrict__ pair_idx, // (B, N, 2) int32
    float*       __restrict__ out)      // (B, 2) fp32
{
    // B operand pre-packed per 16-wide P-tile in WMMA 4x16 f32 register layout:
    // tile t, lane l  ->  s_b[t*64 + l*2 + {0,1}]
    //   lanes 0..15 : (uvx[p], uvy[p])   rows K=0,1  (p = t*16 + l)
    //   lanes 16..31: (-c[p],  0.0f)     rows K=2,3  (p = t*16 + l-16)
    __shared__ float s_b[NTILES * 64];   // 64 KB
    __shared__ float s_Yx[NPAIR];        // intersection row coord
    __shared__ float s_Yy[NPAIR];        // intersection col coord
    __shared__ float s_w[NPAIR];         // vote counts
    __shared__ float s_rx[NPAIR];
    __shared__ float s_ry[NPAIR];

    const int b   = blockIdx.x;
    const int tid = threadIdx.x;

    const float* uv_c0 = uv_img + (size_t)b * CH * IMH * IMW;
    const float* uv_c1 = uv_c0 + IMH * IMW;
    const int*   ptsb  = pts      + (size_t)b * NPTS * 2;
    const int*   pib   = pair_idx + (size_t)b * NPAIR * 2;

    // ---- Phase 1a: gather flow vectors, pack B tiles in WMMA layout ----
    for (int p = tid; p < NPTS; p += THREADS) {
        int pr = ptsb[2 * p + 0];            // row  (== pts[:,0])
        int pc = ptsb[2 * p + 1];            // col  (== pts[:,1])
        int off = pr * IMW + pc;
        float ux = uv_c0[off];
        float uy = uv_c1[off];
        float nc = -((float)pr * ux + (float)pc * uy);
        int t = p >> 4;           // P-tile
        int s = p & 15;           // slot within tile
        *(float2*)&s_b[t * 64 + s * 2]            = make_float2(ux, uy);  // lanes 0-15
        *(float2*)&s_b[t * 64 + (s + 16) * 2]     = make_float2(nc, 0.0f); // lanes 16-31
    }

    // ---- Phase 1b: per-pair line intersection Y (one pair / thread) ----
    {
        int n  = tid;
        int i0 = pib[2 * n + 0];
        int i1 = pib[2 * n + 1];
        int p0r = ptsb[2 * i0 + 0], p0c = ptsb[2 * i0 + 1];
        int p1r = ptsb[2 * i1 + 0], p1c = ptsb[2 * i1 + 1];
        int o0 = p0r * IMW + p0c;
        int o1 = p1r * IMW + p1c;
        float u0x = uv_c0[o0], u0y = uv_c1[o0];
        float u1x = uv_c0[o1], u1y = uv_c1[o1];
        // A = [[u0x, -u1x], [u0y, -u1y]] ; Bv = pt1 - pt0 ; X = A^-1 Bv
        float b0 = (float)(p1r - p0r);
        float b1 = (float)(p1c - p0c);
        float det = u1x * u0y - u0x * u1y;
        float x0 = (-u1y * b0 + u1x * b1) / det;  // pinv ~= inv (a.s. nonsingular)
        s_Yx[n] = x0 * u0x + (float)p0r;
        s_Yy[n] = x0 * u0y + (float)p0c;
    }
    __syncthreads();

    // ---- Phase 2: vote counts via WMMA ----
    // S(256x4096) = [Yx Yy 1] (256x3) @ [uvx; uvy; -c] (3x4096); w[n] = #{p : S[n,p] > 0}
    // V_WMMA_F32_16X16X4_F32, K padded 3 -> 4 with zeros.
    const int  lane = tid & 31;
    const int  wave = tid >> 5;
    const int  col  = lane & 15;
    const bool hi   = lane >= 16;   // high half of wave holds K=2,3 operand rows

    for (int k = 0; k < 2; ++k) {
        const int mt = wave * 2 + k;     // 8 waves x 2 = 16 M-tiles
        const int m0 = mt * 16;

        // A tile 16x4 fp32 (2 VGPRs): lanes 0-15 carry K=0 (Yx), K=1 (Yy);
        // lanes 16-31 carry K=2 (constant 1), K=3 (zero pad).
        v2f a;
        a.x = hi ? 1.0f : s_Yx[m0 + col];
        a.y = hi ? 0.0f : s_Yy[m0 + col];

        int cnt[8] = {0, 0, 0, 0, 0, 0, 0, 0};

        const float* bp = &s_b[lane * 2];
        #pragma unroll 4
        for (int pt = 0; pt < NTILES; ++pt) {
            // One aligned ds_load_b64: this lane's 2 B-operand VGPRs for tile pt.
            v2f bb = *(const v2f*)(bp + pt * 64);

            v8f c = {};  // no accumulate: D = A*B
            v8f d = __builtin_amdgcn_wmma_f32_16x16x4_f32(
                false, a, false, bb, (short)0, c, false, false);

            #pragma unroll
            for (int v = 0; v < 8; ++v)
                cnt[v] += (d[v] > 0.0f) ? 1 : 0;
        }

        // C/D layout: lane l, VGPR v holds S[m0 + v + 8*(l>=16)][pt*16 + l%16].
        // Reduce across the 16 lanes of each half (xor masks stay within halves).
        #pragma unroll
        for (int v = 0; v < 8; ++v) {
            int c = cnt[v];
            c += __shfl_xor(c, 1, 32);
            c += __shfl_xor(c, 2, 32);
            c += __shfl_xor(c, 4, 32);
            c += __shfl_xor(c, 8, 32);
            if (col == 0)
                s_w[m0 + v + (hi ? 8 : 0)] = (float)c;
        }
    }
    __syncthreads();

    // ---- Phase 3: deterministic weighted-mean reduction ----
    {
        float w = s_w[tid];
        s_rx[tid] = s_Yx[tid] * w;
        s_ry[tid] = s_Yy[tid] * w;
    }
    __syncthreads();
    for (int s = NPAIR / 2; s > 0; s >>= 1) {
        if (tid < s) {
            s_rx[tid] += s_rx[tid + s];
            s_ry[tid] += s_ry[tid + s];
            s_w[tid]  += s_w[tid + s];
        }
        __syncthreads();
    }
    if (tid == 0) {
        float inv_sw = 1.0f / s_w[0];
        // weighted_mean[::-1] : col-coordinate first, then row-coordinate
        out[2 * b + 0] = s_ry[0] * inv_sw;
        out[2 * b + 1] = s_rx[0] * inv_sw;
    }
}

extern "C" void kernel_launch(void* const* d_in, const int* in_sizes, int n_in,
                              void* d_out, int out_size, void* d_ws, size_t ws_size,
                              hipStream_t stream) {
    const float* uv_img   = (const float*)d_in[0];
    const int*   pts      = (const int*)d_in[1];
    const int*   pair_idx = (const int*)d_in[2];
    float*       out      = (float*)d_out;
    (void)in_sizes; (void)n_in; (void)out_size; (void)d_ws; (void)ws_size;

    hough_voting_kernel<<<BATCH, THREADS, 0, stream>>>(uv_img, pts, pair_idx, out);
}